// Granularity_Fusion_29515015258378
// MI455X (gfx1250) — compile-verified
//
#include <hip/hip_runtime.h>
#include <math.h>

// ---------------------------------------------------------------------------
// Granularity_Fusion on MI455X (gfx1250).
// Exact-f32 pipeline; irreducible GEMMs run on V_WMMA_F32_16X16X4_F32.
// ---------------------------------------------------------------------------

typedef __attribute__((ext_vector_type(2))) float v2f;
typedef __attribute__((ext_vector_type(8))) float v8f;

#define DIMC   192
#define NHEADS 4
#define IMG    48
#define HW1    2304           // 48*48
#define SCALE_ATT 0.14433756729740643f   // 48^-0.5
#define INV_SQRT2 0.70710678118654752f

// ===========================================================================
// FSA stage 1 kernel: WMMA GEMM  xn = xs @ conv_w^T + conv_b  fused with
// LayerNorm + exact GELU + q/k projections.
// block = 384 threads (12 waves, one 16-col tile each), grid = B*N/16.
// ===========================================================================
__global__ __launch_bounds__(384)
void fsa_main(const float* __restrict__ xin,     // [B*N, 192] (raw BCHW buffer)
              const float* __restrict__ conv_w,  // [192,192]
              const float* __restrict__ conv_b,  // [192]
              const float* __restrict__ ln_g, const float* __restrict__ ln_b,
              const float* __restrict__ q_w,     // [4,192]
              const float* __restrict__ k_w,     // [4,192]
              float* __restrict__ xa,            // [B*N, 192]
              float* __restrict__ pm,            // [B,4,N]
              int N)
{
    __shared__ float xn[16 * DIMC];              // 12 KB
    const int tile = blockIdx.x;
    const int lane = threadIdx.x & 31;
    const int wave = threadIdx.x >> 5;           // 0..11 -> 16-col block
    const int hf   = lane >> 4;                  // 0/1
    const int l    = lane & 15;
    const int row0 = tile * 16;                  // flattened row (b*N + n)

    v8f acc = {};
    const float* ap = xin    + (size_t)(row0 + l)       * DIMC + 2 * hf;
    const float* wp = conv_w + (size_t)(wave * 16 + l)  * DIMC + 2 * hf;
    for (int k = 0; k < DIMC; k += 4) {
        v2f av, bv;
        av[0] = ap[k]; av[1] = ap[k + 1];
        bv[0] = wp[k]; bv[1] = wp[k + 1];
        acc = __builtin_amdgcn_wmma_f32_16x16x4_f32(false, av, false, bv,
                                                    (short)0, acc, false, false);
    }
    const int col = wave * 16 + l;
    const float bias = conv_b[col];
    for (int r = 0; r < 8; ++r)
        xn[(r + 8 * hf) * DIMC + col] = acc[r] + bias;
    __syncthreads();

    // Epilogue: LN + GELU + q/k.  16 rows, one thread each (cheap vs launch).
    if (threadIdx.x < 16) {
        const int m    = threadIdx.x;
        const int grow = row0 + m;               // b*N + n
        const int b    = grow / N;
        const int n    = grow - b * N;
        const float* xrow = &xn[m * DIMC];

        float mean = 0.f;
        for (int c = 0; c < DIMC; ++c) mean += xrow[c];
        mean *= (1.0f / DIMC);
        float var = 0.f;
        for (int c = 0; c < DIMC; ++c) { float d = xrow[c] - mean; var += d * d; }
        var *= (1.0f / DIMC);
        const float rstd = rsqrtf(var + 1e-5f);

        float kk0 = 0.f, kk1 = 0.f, kk2 = 0.f, kk3 = 0.f;
        float* xaout = xa + (size_t)grow * DIMC;
        for (int c = 0; c < DIMC; ++c) {
            float h = (xrow[c] - mean) * rstd * ln_g[c] + ln_b[c];
            float g = 0.5f * h * (1.0f + erff(h * INV_SQRT2));   // exact GELU
            xaout[c] = g;
            kk0 += k_w[0 * DIMC + c] * g;
            kk1 += k_w[1 * DIMC + c] * g;
            kk2 += k_w[2 * DIMC + c] * g;
            kk3 += k_w[3 * DIMC + c] * g;
        }
        float q = 0.f;
        const float* xsrow = xin + (size_t)grow * DIMC;
        for (int c = 0; c < DIMC; ++c)
            q += (q_w[c] + q_w[DIMC + c] + q_w[2 * DIMC + c] + q_w[3 * DIMC + c]) * xsrow[c];

        float* pmb = pm + (size_t)b * NHEADS * N + n;
        pmb[0 * N] = q * kk0;
        pmb[1 * N] = q * kk1;
        pmb[2 * N] = q * kk2;
        pmb[3 * N] = q * kk3;
    }
}

// ===========================================================================
// Per-head softmax of the bilinear-upsampled pair-averaged pm.
// grid = B*NH, block = 256.
// ===========================================================================
__global__ __launch_bounds__(256)
void fsa_softmax(const float* __restrict__ pm, float* __restrict__ abuf, int N)
{
    __shared__ float pp[HW1 / 2];    // pair-averaged pm (max 1152)
    __shared__ float red[256];
    const int bh = blockIdx.x;
    const float* p = pm + (size_t)bh * N;
    float* aout = abuf + (size_t)bh * N;
    const int Nh = N / 2;

    for (int m = threadIdx.x; m < Nh; m += 256)
        pp[m] = 0.5f * (p[2 * m] + p[2 * m + 1]);
    __syncthreads();

    auto uf = [&](int j) -> float {      // half-pixel bilinear x2 (edge clamp)
        int m = j >> 1;
        if ((j & 1) == 0) { int i0 = (m > 0) ? m - 1 : 0;       return 0.25f * pp[i0] + 0.75f * pp[m]; }
        else              { int i1 = (m < Nh - 1) ? m + 1 : m;  return 0.75f * pp[m]  + 0.25f * pp[i1]; }
    };

    float mx = -3.4e38f;
    for (int j = threadIdx.x; j < N; j += 256) mx = fmaxf(mx, SCALE_ATT * uf(j));
    red[threadIdx.x] = mx; __syncthreads();
    for (int s = 128; s > 0; s >>= 1) {
        if (threadIdx.x < s) red[threadIdx.x] = fmaxf(red[threadIdx.x], red[threadIdx.x + s]);
        __syncthreads();
    }
    mx = red[0]; __syncthreads();

    float sm = 0.f;
    for (int j = threadIdx.x; j < N; j += 256) sm += expf(SCALE_ATT * uf(j) - mx);
    red[threadIdx.x] = sm; __syncthreads();
    for (int s = 128; s > 0; s >>= 1) {
        if (threadIdx.x < s) red[threadIdx.x] += red[threadIdx.x + s];
        __syncthreads();
    }
    const float inv = 1.0f / red[0];

    for (int j = threadIdx.x; j < N; j += 256)
        aout[j] = expf(SCALE_ATT * uf(j) - mx) * inv;
}

__global__ void fsa_wsum(const float* __restrict__ abuf, float* __restrict__ w, int N)
{
    const int b = blockIdx.x;
    const float* a = abuf + (size_t)b * NHEADS * N;
    for (int j = threadIdx.x; j < N; j += 256)
        w[b * N + j] = a[j] + a[N + j] + a[2 * N + j] + a[3 * N + j];
}

__global__ void fsa_xw(const float* __restrict__ xa, const float* __restrict__ w,
                       float* __restrict__ xw, int N)
{
    const int b = blockIdx.x, c = threadIdx.x;    // block = 192
    const float* xab = xa + (size_t)b * N * DIMC;
    const float* wb  = w  + (size_t)b * N;
    float acc = 0.f;
    for (int n = 0; n < N; ++n) acc += wb[n] * xab[(size_t)n * DIMC + c];
    xw[b * DIMC + c] = acc;
}

__global__ void fsa_out(const float* __restrict__ xw, const float* __restrict__ v_w,
                        const float* __restrict__ proj_w, const float* __restrict__ proj_b,
                        float* __restrict__ rout)
{
    __shared__ float xl[DIMC], sl[DIMC];
    const int b = blockIdx.x, t = threadIdx.x;    // block = 192
    xl[t] = xw[b * DIMC + t];
    __syncthreads();
    float s = 0.f;
    for (int c = 0; c < DIMC; ++c) s += v_w[t * DIMC + c] * xl[c];
    sl[t] = s;
    __syncthreads();
    float r = proj_b[t];
    for (int c = 0; c < DIMC; ++c) r += proj_w[t * DIMC + c] * sl[c];
    rout[b * DIMC + t] = r;
}

// ===========================================================================
// avgpool2 of the tiled-r FSA output: attn[b,c,h,w] = r[b,(c*HW+h*W+w)%192]
// out: [B,192,H/2,W/2]
// ===========================================================================
__global__ void tile_pool(const float* __restrict__ r, float* __restrict__ out,
                          int H, int W)
{
    const int Ho = H / 2, Wo = W / 2, HW = H * W;
    const int total = 2 * DIMC * Ho * Wo;
    int i = blockIdx.x * 256 + threadIdx.x;
    if (i >= total) return;
    int wo = i % Wo, t = i / Wo;
    int ho = t % Ho; t /= Ho;
    int c = t % DIMC; int b = t / DIMC;
    const float* rb = r + b * DIMC;
    const int h0 = 2 * ho, w0 = 2 * wo;
    float s = rb[(c * HW + h0 * W + w0) % DIMC]
            + rb[(c * HW + h0 * W + w0 + 1) % DIMC]
            + rb[(c * HW + (h0 + 1) * W + w0) % DIMC]
            + rb[(c * HW + (h0 + 1) * W + w0 + 1) % DIMC];
    out[i] = 0.25f * s;
}

// attnT[b, sp(48x48), c] = r[b, (c*HWs + sps)%192], sps = (h>>sh)*Ws + (w>>sh)
__global__ void make_attnT(const float* __restrict__ r, float* __restrict__ out,
                           int HWs, int Ws, int sh)
{
    const int total = 2 * HW1 * DIMC;
    int i = blockIdx.x * 256 + threadIdx.x;
    if (i >= total) return;
    int c = i % DIMC; int t = i / DIMC;
    int sp = t % HW1; int b = t / HW1;
    int h = sp / IMG, w = sp % IMG;
    int sps = (h >> sh) * Ws + (w >> sh);
    out[i] = r[b * DIMC + (c * HWs + sps) % DIMC];
}

// ===========================================================================
// Generic WMMA f32 GEMM: Out[m,n] = sum_k A[m,k]*W[n,k] + bias[n]
// A: [(B*M), K] row-major; W: [NC, K] row-major.
// grid = dim3(B*M/16, NC/16), block = 32 (one wave per tile).
// ===========================================================================
__global__ __launch_bounds__(32)
void wmma_gemm_bias(const float* __restrict__ A, const float* __restrict__ W,
                    const float* __restrict__ bias, float* __restrict__ Out,
                    int K, int NC)
{
    const int rowTile = blockIdx.x, colTile = blockIdx.y;
    const int lane = threadIdx.x;
    const int hf = lane >> 4, l = lane & 15;
    v8f acc = {};
    const float* ap = A + (size_t)(rowTile * 16 + l) * K + 2 * hf;
    const float* wp = W + (size_t)(colTile * 16 + l) * K + 2 * hf;
    for (int k = 0; k < K; k += 4) {
        v2f av, bv;
        av[0] = ap[k]; av[1] = ap[k + 1];
        bv[0] = wp[k]; bv[1] = wp[k + 1];
        acc = __builtin_amdgcn_wmma_f32_16x16x4_f32(false, av, false, bv,
                                                    (short)0, acc, false, false);
    }
    const int col = colTile * 16 + l;
    const float bv = bias ? bias[col] : 0.f;
    for (int r = 0; r < 8; ++r) {
        int row = rowTile * 16 + r + 8 * hf;
        Out[(size_t)row * NC + col] = acc[r] + bv;
    }
}

// SA = [mean, max] over the 192 concatenated U channels. UiT: [B,2304,64].
__global__ void sa_kernel(const float* __restrict__ U1, const float* __restrict__ U2,
                          const float* __restrict__ U3, float* __restrict__ SA)
{
    const int total = 2 * HW1;
    int i = blockIdx.x * 256 + threadIdx.x;
    if (i >= total) return;
    int sp = i % HW1, b = i / HW1;
    const float* u1 = U1 + (size_t)i * 64;
    const float* u2 = U2 + (size_t)i * 64;
    const float* u3 = U3 + (size_t)i * 64;
    float s = 0.f, mx = -3.4e38f;
    for (int o = 0; o < 64; ++o) { float v = u1[o]; s += v; mx = fmaxf(mx, v); }
    for (int o = 0; o < 64; ++o) { float v = u2[o]; s += v; mx = fmaxf(mx, v); }
    for (int o = 0; o < 64; ++o) { float v = u3[o]; s += v; mx = fmaxf(mx, v); }
    SA[(b * 2 + 0) * HW1 + sp] = s * (1.0f / 192.0f);
    SA[(b * 2 + 1) * HW1 + sp] = mx;
}

// mask = sigmoid(conv7x7(SA) + b), pad 3. mask: [B,3,2304]
__global__ void mask_kernel(const float* __restrict__ SA, const float* __restrict__ w7,
                            const float* __restrict__ b7, float* __restrict__ mask)
{
    const int total = 2 * HW1;
    int i = blockIdx.x * 256 + threadIdx.x;
    if (i >= total) return;
    int sp = i % HW1, b = i / HW1;
    int h = sp / IMG, w = sp % IMG;
    for (int m = 0; m < 3; ++m) {
        float acc = b7[m];
        for (int ci = 0; ci < 2; ++ci)
            for (int kh = 0; kh < 7; ++kh) {
                int hh = h + kh - 3; if (hh < 0 || hh >= IMG) continue;
                for (int kw = 0; kw < 7; ++kw) {
                    int ww = w + kw - 3; if (ww < 0 || ww >= IMG) continue;
                    acc += SA[(b * 2 + ci) * HW1 + hh * IMG + ww]
                         * w7[((m * 2 + ci) * 7 + kh) * 7 + kw];
                }
            }
        mask[(b * 3 + m) * HW1 + sp] = 1.0f / (1.0f + expf(-acc));
    }
}

__global__ void fuse_kernel(const float* __restrict__ U1, const float* __restrict__ U2,
                            const float* __restrict__ U3, const float* __restrict__ mask,
                            float* __restrict__ fused)
{
    const int total = 2 * HW1 * 64;
    int i = blockIdx.x * 256 + threadIdx.x;
    if (i >= total) return;
    int t = i / 64;
    int sp = t % HW1; int b = t / HW1;
    float m0 = mask[(b * 3 + 0) * HW1 + sp];
    float m1 = mask[(b * 3 + 1) * HW1 + sp];
    float m2 = mask[(b * 3 + 2) * HW1 + sp];
    fused[i] = U1[i] * m0 + U2[i] * m1 + U3[i] * m2;
}

// sc[b,c] = mean over 48x48 of (attn2 + attn3), via r2/r3 index formulas.
__global__ void sc_kernel(const float* __restrict__ r2, const float* __restrict__ r3,
                          float* __restrict__ sc)
{
    const int b = blockIdx.x, c = threadIdx.x;   // block = 192
    const float* rb2 = r2 + b * DIMC;
    const float* rb3 = r3 + b * DIMC;
    float acc = 0.f;
    for (int sp = 0; sp < HW1; ++sp) {
        int h = sp / IMG, w = sp % IMG;
        int i2 = ((h >> 1) * 24 + (w >> 1)) % DIMC;               // c*576 % 192 == 0
        int i3 = (c * 144 + (h >> 2) * 12 + (w >> 2)) % DIMC;
        acc += rb2[i2] + rb3[i3];
    }
    sc[b * DIMC + c] = acc * (1.0f / HW1);
}

// CS-module gating: z = sigmoid(fc2(relu(bn(fc1(sc)))))
__global__ void z_kernel(const float* __restrict__ sc, const float* __restrict__ fc1w,
                         const float* __restrict__ fc1b, const float* __restrict__ bng,
                         const float* __restrict__ bnb, const float* __restrict__ fc2w,
                         const float* __restrict__ fc2b, float* __restrict__ z)
{
    __shared__ float scl[DIMC], tl[96];
    const int b = blockIdx.x, t = threadIdx.x;   // block = 192
    scl[t] = sc[b * DIMC + t];
    __syncthreads();
    if (t < 96) {
        float a = fc1b[t];
        for (int c = 0; c < DIMC; ++c) a += fc1w[t * DIMC + c] * scl[c];
        a = a * rsqrtf(1.0f + 1e-5f) * bng[t] + bnb[t];
        tl[t] = fmaxf(a, 0.f);
    }
    __syncthreads();
    float a = fc2b[t];
    for (int j = 0; j < 96; ++j) a += fc2w[t * 96 + j] * tl[j];
    z[b * DIMC + t] = 1.0f / (1.0f + expf(-a));
}

// out = (SpreT + conv2_b)*x + z*attn2 + (1-z)*attn3  (bias already in SpreT)
__global__ void final_kernel(const float* __restrict__ SpreT, const float* __restrict__ x,
                             const float* __restrict__ r2, const float* __restrict__ r3,
                             const float* __restrict__ z, float* __restrict__ out)
{
    const int total = 2 * DIMC * HW1;
    int i = blockIdx.x * 256 + threadIdx.x;
    if (i >= total) return;
    int sp = i % HW1; int t = i / HW1;
    int c = t % DIMC; int b = t / DIMC;
    int h = sp / IMG, w = sp % IMG;
    float a2 = r2[b * DIMC + (((h >> 1) * 24 + (w >> 1)) % DIMC)];
    float a3 = r3[b * DIMC + ((c * 144 + (h >> 2) * 12 + (w >> 2)) % DIMC)];
    float zz = z[b * DIMC + c];
    float ans = zz * a2 + (1.0f - zz) * a3;
    float Sv = SpreT[((size_t)b * HW1 + sp) * DIMC + c] * x[i];
    out[i] = Sv + ans;
}

// ===========================================================================
// Host launcher
// ===========================================================================
extern "C" void kernel_launch(void* const* d_in, const int* in_sizes, int n_in,
                              void* d_out, int out_size, void* d_ws, size_t ws_size,
                              hipStream_t stream)
{
    (void)in_sizes; (void)n_in; (void)out_size; (void)ws_size;
    const float* x = (const float*)d_in[0];
    struct FsaP { const float *qw,*kw,*vw,*pw,*pb,*cw,*cb,*lg,*lb; };
    auto getf = [&](int base) {
        FsaP p;
        p.qw=(const float*)d_in[base+0]; p.kw=(const float*)d_in[base+1];
        p.vw=(const float*)d_in[base+2]; p.pw=(const float*)d_in[base+3];
        p.pb=(const float*)d_in[base+4]; p.cw=(const float*)d_in[base+5];
        p.cb=(const float*)d_in[base+6]; p.lg=(const float*)d_in[base+7];
        p.lb=(const float*)d_in[base+8];
        return p;
    };
    FsaP f1 = getf(1), f2 = getf(10), f3 = getf(19);
    const float* conv0_w = (const float*)d_in[28];
    const float* conv0_b = (const float*)d_in[29];
    const float* conv1_w = (const float*)d_in[30];
    const float* conv1_b = (const float*)d_in[31];
    const float* conv2_w = (const float*)d_in[32];
    const float* conv2_b = (const float*)d_in[33];
    const float* fc1_w   = (const float*)d_in[34];
    const float* fc1_b   = (const float*)d_in[35];
    const float* fc2_w   = (const float*)d_in[36];
    const float* fc2_b   = (const float*)d_in[37];
    const float* bn_g    = (const float*)d_in[38];
    const float* bn_b    = (const float*)d_in[39];

    const int N1 = 2304, N2 = 576, N3 = 144;

    // workspace carving (floats)
    float* p = (float*)d_ws;
    auto take = [&](size_t n) { float* q = p; p += n; return q; };
    float* xa1     = take((size_t)2 * N1 * DIMC);
    float* xa2     = take((size_t)2 * N2 * DIMC);
    float* xa3     = take((size_t)2 * N3 * DIMC);
    float* pm1     = take((size_t)2 * NHEADS * N1);
    float* pm2     = take((size_t)2 * NHEADS * N2);
    float* pm3     = take((size_t)2 * NHEADS * N3);
    float* ab1     = take((size_t)2 * NHEADS * N1);
    float* ab2     = take((size_t)2 * NHEADS * N2);
    float* ab3     = take((size_t)2 * NHEADS * N3);
    float* w1      = take((size_t)2 * N1);
    float* w2      = take((size_t)2 * N2);
    float* w3      = take((size_t)2 * N3);
    float* xwb     = take((size_t)2 * DIMC);
    float* r1      = take((size_t)2 * DIMC);
    float* r2      = take((size_t)2 * DIMC);
    float* r3      = take((size_t)2 * DIMC);
    float* pooled1 = take((size_t)2 * DIMC * 576);
    float* pooled2 = take((size_t)2 * DIMC * 144);
    float* attnT   = take((size_t)2 * HW1 * DIMC);   // reused for attnT1/2/3
    float* U1T     = take((size_t)2 * HW1 * 64);
    float* U2T     = take((size_t)2 * HW1 * 64);
    float* U3T     = take((size_t)2 * HW1 * 64);
    float* SA      = take((size_t)2 * 2 * HW1);
    float* maskb   = take((size_t)2 * 3 * HW1);
    float* fusedT  = take((size_t)2 * HW1 * 64);
    float* SpreT   = take((size_t)2 * HW1 * DIMC);
    float* scb     = take((size_t)2 * DIMC);
    float* zb      = take((size_t)2 * DIMC);

    auto run_fsa = [&](const float* xin, const FsaP& pr, int N,
                       float* xa, float* pm, float* ab, float* w, float* r) {
        fsa_main<<<dim3(2 * N / 16), dim3(384), 0, stream>>>(
            xin, pr.cw, pr.cb, pr.lg, pr.lb, pr.qw, pr.kw, xa, pm, N);
        fsa_softmax<<<dim3(2 * NHEADS), dim3(256), 0, stream>>>(pm, ab, N);
        fsa_wsum<<<dim3(2), dim3(256), 0, stream>>>(ab, w, N);
        fsa_xw<<<dim3(2), dim3(DIMC), 0, stream>>>(xa, w, xwb, N);
        fsa_out<<<dim3(2), dim3(DIMC), 0, stream>>>(xwb, pr.vw, pr.pw, pr.pb, r);
    };

    // --- three FSA stages -------------------------------------------------
    run_fsa(x, f1, N1, xa1, pm1, ab1, w1, r1);
    tile_pool<<<dim3((2 * DIMC * 576 + 255) / 256), dim3(256), 0, stream>>>(r1, pooled1, 48, 48);
    run_fsa(pooled1, f2, N2, xa2, pm2, ab2, w2, r2);
    tile_pool<<<dim3((2 * DIMC * 144 + 255) / 256), dim3(256), 0, stream>>>(r2, pooled2, 24, 24);
    run_fsa(pooled2, f3, N3, xa3, pm3, ab3, w3, r3);

    // --- U_i = conv0(attn_i) as WMMA GEMMs on [HW,192] layouts ------------
    const int elemT = 2 * HW1 * DIMC;
    make_attnT<<<dim3((elemT + 255) / 256), dim3(256), 0, stream>>>(r1, attnT, 2304, 48, 0);
    wmma_gemm_bias<<<dim3(2 * HW1 / 16, 64 / 16), dim3(32), 0, stream>>>(
        attnT, conv0_w, conv0_b, U1T, DIMC, 64);
    make_attnT<<<dim3((elemT + 255) / 256), dim3(256), 0, stream>>>(r2, attnT, 576, 24, 1);
    wmma_gemm_bias<<<dim3(2 * HW1 / 16, 64 / 16), dim3(32), 0, stream>>>(
        attnT, conv0_w, conv0_b, U2T, DIMC, 64);
    make_attnT<<<dim3((elemT + 255) / 256), dim3(256), 0, stream>>>(r3, attnT, 144, 12, 2);
    wmma_gemm_bias<<<dim3(2 * HW1 / 16, 64 / 16), dim3(32), 0, stream>>>(
        attnT, conv0_w, conv0_b, U3T, DIMC, 64);

    // --- spatial attention mask & fusion ----------------------------------
    sa_kernel<<<dim3((2 * HW1 + 255) / 256), dim3(256), 0, stream>>>(U1T, U2T, U3T, SA);
    mask_kernel<<<dim3((2 * HW1 + 255) / 256), dim3(256), 0, stream>>>(SA, conv1_w, conv1_b, maskb);
    fuse_kernel<<<dim3((2 * HW1 * 64 + 255) / 256), dim3(256), 0, stream>>>(U1T, U2T, U3T, maskb, fusedT);
    wmma_gemm_bias<<<dim3(2 * HW1 / 16, DIMC / 16), dim3(32), 0, stream>>>(
        fusedT, conv2_w, conv2_b, SpreT, 64, DIMC);

    // --- CS module + final combine ----------------------------------------
    sc_kernel<<<dim3(2), dim3(DIMC), 0, stream>>>(r2, r3, scb);
    z_kernel<<<dim3(2), dim3(DIMC), 0, stream>>>(scb, fc1_w, fc1_b, bn_g, bn_b, fc2_w, fc2_b, zb);
    final_kernel<<<dim3((2 * DIMC * HW1 + 255) / 256), dim3(256), 0, stream>>>(
        SpreT, x, r2, r3, zb, (float*)d_out);
}